// MaxAggregator_42442866819640
// MI455X (gfx1250) — compile-verified
//
#include <hip/hip_runtime.h>

// Hypergraph max-aggregation for MI455X (gfx1250, wave32).
//   out[n, :] = max over incidences e with segment_ids[e]==n of table[edge_hyperedges[e], :]
//
// Design (roofline): table = 100K x 128 x 4B = 51.2 MB -> resident in the 192 MB
// global L2, so the 800K row-gathers (410 MB logical) are L2 hits; HBM sees
// ~85 MB total -> a few microseconds at 23.3 TB/s. The limiter is L2 gather
// latency, so: one wave32 per node (lane l owns columns [4l,4l+4) = one b128
// load, 512B/row per wave, fully coalesced), 50K waves for occupancy, 4-deep
// unroll with independent accumulators + global_prefetch_b8 ahead for MLP,
// and non-temporal b128 stores so the 25.6 MB output never evicts the table.

typedef float floatx4 __attribute__((ext_vector_type(4)));

static constexpr int D    = 128; // embedding dim
static constexpr int WAVE = 32;  // gfx1250 is wave32

// Wave-cooperative 32-ary lower_bound over sorted int array in [lo0, n].
// All 32 lanes must be active. Invariant: answer (first idx with
// seg[idx] >= target) lies in [lo, hi]; caller guarantees answer >= lo0.
__device__ __forceinline__ int lower_bound_wave(const int* __restrict__ seg,
                                                int lo0, int n, int target,
                                                int lane) {
  int lo = lo0, hi = n;
  while (hi - lo > WAVE) {
    const int len   = hi - lo;
    const int chunk = (len + WAVE - 1) >> 5;        // ceil(len/32)
    const int q     = lo + lane * chunk;            // chunk starts
    const bool c    = (q < hi) && (seg[q] < target);
    const unsigned m = __builtin_amdgcn_ballot_w32(c);
    const int k = __builtin_popcount(m);            // prefix of true lanes
    if (k == 0) { hi = lo; break; }                 // answer == lo
    const int nlo = lo + (k - 1) * chunk + 1;
    const int nhi = lo + k * chunk;
    lo = nlo;
    hi = (nhi < hi) ? nhi : hi;                     // new span <= chunk-1
  }
  // Final linear sweep, <=32 candidates, one ballot.
  const int  q = lo + lane;
  const bool c = (q < hi) && (seg[q] < target);
  lo += __builtin_popcount(__builtin_amdgcn_ballot_w32(c));
  return lo;
}

__device__ __forceinline__ void fmax4(floatx4& a, const floatx4 v) {
  a.x = fmaxf(a.x, v.x);
  a.y = fmaxf(a.y, v.y);
  a.z = fmaxf(a.z, v.z);
  a.w = fmaxf(a.w, v.w);
}

__global__ __launch_bounds__(256) void hyperedge_segmax_kernel(
    const float* __restrict__ table,   // [100000, 128]
    const int*   __restrict__ edge_h,  // [E] hyperedge id per incidence
    const int*   __restrict__ seg,     // [E] sorted owning node per incidence
    float*       __restrict__ out,     // [n_nodes, 128]
    int n_edges, int n_nodes) {
  const int lane = (int)(threadIdx.x & (WAVE - 1));
  const int node = (int)((blockIdx.x * blockDim.x + threadIdx.x) >> 5);
  if (node >= n_nodes) return;  // wave-uniform: whole wave exits together

  // CSR range of this node via two wave-cooperative searches; the second
  // search starts at `beg` (end >= beg), saving a probe round.
  const int beg = lower_bound_wave(seg, 0,   n_edges, node,     lane);
  const int end = lower_bound_wave(seg, beg, n_edges, node + 1, lane);

  const int   col  = lane * 4;
  const float ninf = -__builtin_inff();
  floatx4 a0 = { ninf, ninf, ninf, ninf };
  floatx4 a1 = a0, a2 = a0, a3 = a0;

  int e = beg;
  // Unroll x4 with independent accumulators: 4 b128 gathers in flight per
  // iteration, plus global_prefetch_b8 for the 4 rows after that.
  for (; e + 3 < end; e += 4) {
    const long long h0 = (long long)edge_h[e];
    const long long h1 = (long long)edge_h[e + 1];
    const long long h2 = (long long)edge_h[e + 2];
    const long long h3 = (long long)edge_h[e + 3];
    if (e + 7 < end) {
      __builtin_prefetch(table + (long long)edge_h[e + 4] * D + col, 0, 1);
      __builtin_prefetch(table + (long long)edge_h[e + 5] * D + col, 0, 1);
      __builtin_prefetch(table + (long long)edge_h[e + 6] * D + col, 0, 1);
      __builtin_prefetch(table + (long long)edge_h[e + 7] * D + col, 0, 1);
    }
    const floatx4 v0 = *(const floatx4*)(table + h0 * D + col);
    const floatx4 v1 = *(const floatx4*)(table + h1 * D + col);
    const floatx4 v2 = *(const floatx4*)(table + h2 * D + col);
    const floatx4 v3 = *(const floatx4*)(table + h3 * D + col);
    fmax4(a0, v0);
    fmax4(a1, v1);
    fmax4(a2, v2);
    fmax4(a3, v3);
  }
  for (; e < end; ++e) {
    const long long h = (long long)edge_h[e];
    const floatx4 v = *(const floatx4*)(table + h * D + col);
    fmax4(a0, v);
  }
  fmax4(a0, a1);
  fmax4(a2, a3);
  fmax4(a0, a2);

  // Empty segment -> reference zeros it (segment_max yields -inf there).
  if (end == beg) a0 = (floatx4){ 0.f, 0.f, 0.f, 0.f };

  // Non-temporal b128 store: written once, never re-read; keep L2 for the table.
  __builtin_nontemporal_store(a0, (floatx4*)(out + (long long)node * D + col));
}

extern "C" void kernel_launch(void* const* d_in, const int* in_sizes, int n_in,
                              void* d_out, int out_size, void* d_ws, size_t ws_size,
                              hipStream_t stream) {
  (void)n_in; (void)d_ws; (void)ws_size;
  const float* table  = (const float*)d_in[0];
  const int*   edge_h = (const int*)d_in[1];
  const int*   seg    = (const int*)d_in[2];
  float*       out    = (float*)d_out;

  const int n_edges = in_sizes[1];
  const int n_nodes = out_size / D;

  const int waves_per_block = 256 / WAVE;  // 8 waves per block
  const int grid = (n_nodes + waves_per_block - 1) / waves_per_block;

  hipLaunchKernelGGL(hyperedge_segmax_kernel, dim3(grid), dim3(256), 0, stream,
                     table, edge_h, seg, out, n_edges, n_nodes);
}